// SparseGraphConvolution_14491219656730
// MI455X (gfx1250) — compile-verified
//
#include <hip/hip_runtime.h>

typedef __attribute__((ext_vector_type(2))) float v2f;
typedef __attribute__((ext_vector_type(8))) float v8f;

#define T_DIM 128
#define N_DIM 256
#define H_DIM 4

// Low 32 bits of a flat shared-memory address are the LDS byte address
// (CDNA5 ISA 10.2: LDS aperture).
__device__ __forceinline__ unsigned lds_addr32(const void* p) {
  return (unsigned)(unsigned long long)p;
}

// ---------------------------------------------------------------------------
// Stage 0: X1s[t][n][e] = sum_c graph[0][t][n][1+c] * W_s1[e][c]
//          X1t[n][t][e] = sum_c graph[0][t][n][1+c] * W_t1[e][c]
// ---------------------------------------------------------------------------
__global__ __launch_bounds__(256) void x1_kernel(const float* __restrict__ graph,
                                                 const float* __restrict__ Ws1,
                                                 const float* __restrict__ Wt1,
                                                 float* __restrict__ X1s,
                                                 float* __restrict__ X1t) {
  int idx = blockIdx.x * 256 + threadIdx.x;   // over T*N (exact)
  int t = idx >> 8;
  int n = idx & 255;
  const float* gp = graph + ((size_t)(t * N_DIM + n) * 6 + 1);
  float g0 = gp[0], g1 = gp[1], g2 = gp[2], g3 = gp[3], g4 = gp[4];
  float* os = X1s + (size_t)(t * N_DIM + n) * 16;
  float* ot = X1t + (size_t)(n * T_DIM + t) * 16;
#pragma unroll
  for (int e = 0; e < 16; ++e) {
    const float* ws = Ws1 + e * 5;
    const float* wt = Wt1 + e * 5;
    os[e] = ws[0]*g0 + ws[1]*g1 + ws[2]*g2 + ws[3]*g3 + ws[4]*g4;
    ot[e] = wt[0]*g0 + wt[1]*g1 + wt[2]*g2 + wt[3]*g3 + wt[4]*g4;
  }
}

// ---------------------------------------------------------------------------
// Generic batched GCN GEMM:
//   P[m,e'] = prelu( adj[b1,h] (MxM) @ X[b1,h] (Mx16), alpha )
//   Z       = (w2 ? P @ W2^T : P)
//   out[b1*os_b1 + h*os_h + m*os_m + e] = Z[m,e]
// Block = 256 threads = 8 waves; each wave computes one 16x16 tile,
// block covers 128 rows of M.  grid.x = B1 * 4 * (M/128).
// Adjacency chunks are streamed memory->LDS with CDNA5 async copies
// (global_load_async_to_lds_b128, ASYNCcnt) and double-buffered.
// ---------------------------------------------------------------------------
struct GcnP {
  const float* adj;
  const float* x;
  const float* w2;      // 16x16 row-major or nullptr
  const float* alpha;   // 1-element device scalar
  float* out;
  int M;                // adj is MxM (K == M)
  int xs_b1, xs_h;      // X element strides
  int os_b1, os_h, os_m;
};

__global__ __launch_bounds__(256) void gcn_gemm(GcnP p) {
  __shared__ float lds_x[256 * 16];         // X panel, K-pair interleaved
  __shared__ float lds_adj[2][128 * 36];    // double-buffered 128x32 chunk, pad 36
  __shared__ float lds_p[8 * 16 * 18];      // per-wave epilogue tile, stride 18

  const int tid  = threadIdx.x;
  const int lane = tid & 31;
  const int wave = tid >> 5;
  const int M = p.M;
  const int mBlocks = M >> 7;
  int bid = blockIdx.x;
  const int mb = bid % mBlocks;  bid /= mBlocks;
  const int h  = bid & 3;
  const int b1 = bid >> 2;
  const float alpha = *p.alpha;

  const float* adjb = p.adj + (size_t)(b1 * H_DIM + h) * M * M + (size_t)(mb << 7) * M;
  const float* xb   = p.x + (size_t)b1 * p.xs_b1 + (size_t)h * p.xs_h;

  // Preload whole X panel (M x 16), pair-interleaved: lds_x[(k>>1)*32 + 2n + (k&1)]
  for (int i = tid; i < M * 16; i += 256) {
    int k = i >> 4, n = i & 15;
    lds_x[((k >> 1) << 5) + (n << 1) + (k & 1)] = xb[i];
  }

  const int nlo   = lane & 15;
  const int khalf = (lane < 16) ? 0 : 2;  // A/B frag: lane halves hold K+0..1 / K+2..3
  const int mh    = (lane < 16) ? 0 : 8;  // C/D frag: lane halves hold rows r / r+8
  const int r0    = tid >> 3;             // staging: row within 32-row group
  const int cb    = (tid & 7) << 2;       // staging: 4-float column chunk

  // B-fragments of W2^T (B[k][n] = W2[n][k])
  v2f w2f[4];
  if (p.w2) {
#pragma unroll
    for (int j = 0; j < 4; ++j) {
      w2f[j].x = p.w2[nlo * 16 + 4 * j + khalf];
      w2f[j].y = p.w2[nlo * 16 + 4 * j + khalf + 1];
    }
  }

  // Async-copy one 128x32 adjacency chunk into LDS buffer `buf`.
  auto issue_chunk = [&](int k0, int buf) {
#pragma unroll
    for (int rr = 0; rr < 128; rr += 32) {
      const float* src = adjb + (size_t)(r0 + rr) * M + k0 + cb;
      unsigned dst = lds_addr32(&lds_adj[buf][(r0 + rr) * 36 + cb]);
      asm volatile("global_load_async_to_lds_b128 %0, %1, off"
                   :: "v"(dst), "v"(src) : "memory");
      __builtin_prefetch(src + 64, 0, 1);   // speculative, 2 chunks ahead
    }
  };

  v8f c = {};
  const int rowl = (wave << 4) + nlo;
  const int nx2  = (lane < 16) ? (nlo << 1) : (32 + (nlo << 1));

  issue_chunk(0, 0);
  asm volatile("s_wait_asynccnt 0x0" ::: "memory");
  __syncthreads();   // X panel + first chunk visible to all waves

  const int nChunks = M >> 5;
  for (int ch = 0; ch < nChunks; ++ch) {
    const int cur = ch & 1;
    if (ch + 1 < nChunks) issue_chunk((ch + 1) << 5, cur ^ 1);

    const float* la = &lds_adj[cur][rowl * 36];
    const int k0 = ch << 5;
#pragma unroll
    for (int kk = 0; kk < 32; kk += 4) {
      v2f a = *(const v2f*)&la[kk + khalf];
      v2f b = *(const v2f*)&lds_x[(((k0 + kk) >> 1) << 5) + nx2];
      c = __builtin_amdgcn_wmma_f32_16x16x4_f32(false, a, false, b,
                                                (short)0, c, false, false);
    }
    asm volatile("s_wait_asynccnt 0x0" ::: "memory");  // next buffer landed
    __syncthreads();                                   // everyone done with cur
  }

  // PReLU (per-stage alpha)
#pragma unroll
  for (int r = 0; r < 8; ++r) c[r] = (c[r] >= 0.f) ? c[r] : alpha * c[r];

  // Optional fused @W2^T: route accumulator C-layout -> A-layout via LDS
  if (p.w2) {
    float* lp = &lds_p[wave * 16 * 18];
#pragma unroll
    for (int r = 0; r < 8; ++r) lp[(r + mh) * 18 + nlo] = c[r];
    asm volatile("s_wait_dscnt 0x0" ::: "memory");
    v8f z = {};
#pragma unroll
    for (int j = 0; j < 4; ++j) {
      v2f a = *(const v2f*)&lp[nlo * 18 + 4 * j + khalf];
      z = __builtin_amdgcn_wmma_f32_16x16x4_f32(false, a, false, w2f[j],
                                                (short)0, z, false, false);
    }
    c = z;
  }

  // Store with parameterized strides (absorbs all reference transposes)
  {
    const int m0 = (mb << 7) + (wave << 4);
    float* outp = p.out + (size_t)b1 * p.os_b1 + (size_t)h * p.os_h + nlo;
#pragma unroll
    for (int r = 0; r < 8; ++r)
      outp[(size_t)(m0 + r + mh) * p.os_m] = c[r];
  }
}

// ---------------------------------------------------------------------------
// Final: comb=[Hs|Ht] (32) -> tanh(Wm1 comb + bm1) (64) -> softmax(Wm2 h + bm2)
// -> out = w0*Hs + w1*Ht.  One thread per (n,t,h) position.
// ---------------------------------------------------------------------------
__global__ __launch_bounds__(256) void mlp_kernel(const float* __restrict__ Hs,
                                                  const float* __restrict__ Ht,
                                                  const float* __restrict__ Wm1,
                                                  const float* __restrict__ bm1,
                                                  const float* __restrict__ Wm2,
                                                  const float* __restrict__ bm2,
                                                  float* __restrict__ out) {
  __shared__ float sW1[64 * 32];
  __shared__ float sb1[64];
  __shared__ float sW2[2 * 64];
  __shared__ float sb2[2];
  const int tid = threadIdx.x;
  for (int i = tid; i < 2048; i += 256) sW1[i] = Wm1[i];
  if (tid < 64)  sb1[tid] = bm1[tid];
  if (tid < 128) sW2[tid] = Wm2[tid];
  if (tid < 2)   sb2[tid] = bm2[tid];
  __syncthreads();

  const int pidx = blockIdx.x * 256 + tid;   // over N*T*H (exact: 131072)
  const float* hs = Hs + (size_t)pidx * 16;
  const float* ht = Ht + (size_t)pidx * 16;
  float hv[16], tv[16];
#pragma unroll
  for (int e = 0; e < 16; ++e) { hv[e] = hs[e]; tv[e] = ht[e]; }

  float d0 = sb2[0], d1 = sb2[1];
  for (int j = 0; j < 64; ++j) {
    const float* w = sW1 + j * 32;
    float a = sb1[j];
#pragma unroll
    for (int i = 0; i < 16; ++i) a += w[i] * hv[i];
#pragma unroll
    for (int i = 0; i < 16; ++i) a += w[16 + i] * tv[i];
    a = tanhf(a);
    d0 += sW2[j] * a;
    d1 += sW2[64 + j] * a;
  }
  float mx = fmaxf(d0, d1);
  float e0 = __expf(d0 - mx), e1 = __expf(d1 - mx);
  float w0 = e0 / (e0 + e1), w1 = 1.f - w0;

  float* op = out + (size_t)pidx * 16;
#pragma unroll
  for (int e = 0; e < 16; ++e) op[e] = w0 * hv[e] + w1 * tv[e];
}

// ---------------------------------------------------------------------------
extern "C" void kernel_launch(void* const* d_in, const int* in_sizes, int n_in,
                              void* d_out, int out_size, void* d_ws, size_t ws_size,
                              hipStream_t stream) {
  const float* graph = (const float*)d_in[0];
  const float* adj_s = (const float*)d_in[1];
  const float* adj_t = (const float*)d_in[2];
  const float* W_s1  = (const float*)d_in[3];
  const float* a_s1  = (const float*)d_in[4];
  const float* W_s2  = (const float*)d_in[5];
  const float* a_s2  = (const float*)d_in[6];
  const float* W_t1  = (const float*)d_in[7];
  const float* a_t1  = (const float*)d_in[8];
  const float* W_t2  = (const float*)d_in[9];
  const float* a_t2  = (const float*)d_in[10];
  const float* Wm1   = (const float*)d_in[11];
  const float* bm1   = (const float*)d_in[12];
  const float* Wm2   = (const float*)d_in[13];
  const float* bm2   = (const float*)d_in[14];

  float* ws  = (float*)d_ws;
  float* X1s = ws;                    // [t][n][16]        524288
  float* X1t = X1s + 524288;          // [n][t][16]        524288
  float* Z2s = X1t + 524288;          // [n][h][t][16]    2097152
  float* Z2t = Z2s + 2097152;         // [t][h][n][16]    2097152
  float* Hs  = Z2t + 2097152;         // [n][t][h][16]    2097152
  float* Ht  = Hs  + 2097152;         // [n][t][h][16]    2097152

  x1_kernel<<<128, 256, 0, stream>>>(graph, W_s1, W_t1, X1s, X1t);

  GcnP p{};
  // Stage 2s: l1s/Z2s  (batch b1=t, M=N=256)
  p.adj = adj_s; p.x = X1s; p.w2 = W_s2; p.alpha = a_s1; p.out = Z2s;
  p.M = 256; p.xs_b1 = 256 * 16; p.xs_h = 0;
  p.os_b1 = 16; p.os_h = T_DIM * 16; p.os_m = H_DIM * T_DIM * 16;
  gcn_gemm<<<1024, 256, 0, stream>>>(p);

  // Stage 4s: Hs  (batch b1=n, M=T=128)
  p.adj = adj_t; p.x = Z2s; p.w2 = nullptr; p.alpha = a_s2; p.out = Hs;
  p.M = 128; p.xs_b1 = H_DIM * T_DIM * 16; p.xs_h = T_DIM * 16;
  p.os_b1 = T_DIM * H_DIM * 16; p.os_h = 16; p.os_m = H_DIM * 16;
  gcn_gemm<<<1024, 256, 0, stream>>>(p);

  // Stage 2t: l1t/Z2t  (batch b1=n, M=T=128)
  p.adj = adj_t; p.x = X1t; p.w2 = W_t2; p.alpha = a_t1; p.out = Z2t;
  p.M = 128; p.xs_b1 = T_DIM * 16; p.xs_h = 0;
  p.os_b1 = 16; p.os_h = N_DIM * 16; p.os_m = H_DIM * N_DIM * 16;
  gcn_gemm<<<1024, 256, 0, stream>>>(p);

  // Stage 4t: Ht  (batch b1=t, M=N=256)
  p.adj = adj_s; p.x = Z2t; p.w2 = nullptr; p.alpha = a_t2; p.out = Ht;
  p.M = 256; p.xs_b1 = H_DIM * N_DIM * 16; p.xs_h = N_DIM * 16;
  p.os_b1 = H_DIM * 16; p.os_h = 16; p.os_m = T_DIM * H_DIM * 16;
  gcn_gemm<<<1024, 256, 0, stream>>>(p);

  mlp_kernel<<<512, 256, 0, stream>>>(Hs, Ht, Wm1, bm1, Wm2, bm2, (float*)d_out);
}